// TimescaleSSM_53068615909652
// MI455X (gfx1250) — compile-verified
//
#include <hip/hip_runtime.h>
#include <math.h>

typedef __attribute__((ext_vector_type(2))) float v2f;
typedef __attribute__((ext_vector_type(8))) float v8f;

#define BATCH 4
#define SEQ 2048
#define LS 1024           // SEQ / TIMESCALE
#define DM 1024
#define DS 64
#define ROWS (BATCH * LS) // 4096
#define SCH 16            // scan timesteps staged per chunk
#define NCHUNK (LS / SCH) // 64

// ---------------------------------------------------------------------------
// c_sum[s] = sum_d Cm[d][s]
// ---------------------------------------------------------------------------
__global__ void csum_kernel(const float* __restrict__ Cm, float* __restrict__ csum) {
    int s = threadIdx.x; // 64 threads
    float acc = 0.f;
    for (int d = 0; d < DM; ++d) acc += Cm[(size_t)d * DS + s];
    csum[s] = acc;
}

// ---------------------------------------------------------------------------
// Subsample (stride 2) + RMSNorm * g_norm  -> x_norm [4096,1024]
// ---------------------------------------------------------------------------
__global__ void rmsnorm_kernel(const float* __restrict__ x, const float* __restrict__ g,
                               float* __restrict__ xn) {
    __shared__ float red[256];
    int r  = blockIdx.x;          // 0..4095
    int b  = r >> 10;
    int ls = r & 1023;
    const float* src = x + ((size_t)b * SEQ + 2 * ls) * DM;
    float ssq = 0.f;
    #pragma unroll
    for (int i = 0; i < 4; ++i) {
        float v = src[threadIdx.x + 256 * i];
        ssq = fmaf(v, v, ssq);
    }
    red[threadIdx.x] = ssq;
    __syncthreads();
    for (int off = 128; off > 0; off >>= 1) {
        if (threadIdx.x < off) red[threadIdx.x] += red[threadIdx.x + off];
        __syncthreads();
    }
    float scale = rsqrtf(red[0] * (1.0f / DM) + 1e-6f);
    float* dst = xn + (size_t)r * DM;
    #pragma unroll
    for (int i = 0; i < 4; ++i) {
        int d = threadIdx.x + 256 * i;
        dst[d] = src[d] * scale * g[d];
    }
}

// ---------------------------------------------------------------------------
// C = A @ W^T + bias via fp32 WMMA 16x16x4. 8 waves/block, wave = 32x32 tile,
// block = 64(M) x 128(N). `repeat` fuses jnp.repeat into the store.
// ---------------------------------------------------------------------------
__global__ void wmma_gemm_kernel(const float* __restrict__ A, const float* __restrict__ W,
                                 const float* __restrict__ bias, float* __restrict__ C,
                                 int K, int ldc, int repeat) {
    const int lane  = threadIdx.x & 31;
    const int wave  = threadIdx.x >> 5;
    const int wm    = wave >> 2;  // 0..1
    const int wn    = wave & 3;   // 0..3
    const int mBase = blockIdx.y * 64 + wm * 32;
    const int nBase = blockIdx.x * 128 + wn * 32;
    const int lhalf = lane >> 4;  // 0/1 -> K pair select
    const int l16   = lane & 15;  // M (for A) / N (for B,C,D)

    v8f acc[2][2] = {};

    for (int kk = 0; kk < K; kk += 4) {
        const int ka = kk + 2 * lhalf;
        v2f afrag[2], bfrag[2];
        #pragma unroll
        for (int i = 0; i < 2; ++i) {
            afrag[i] = *(const v2f*)(A + (size_t)(mBase + i * 16 + l16) * K + ka);
            bfrag[i] = *(const v2f*)(W + (size_t)(nBase + i * 16 + l16) * K + ka);
        }
        #pragma unroll
        for (int i = 0; i < 2; ++i)
            #pragma unroll
            for (int j = 0; j < 2; ++j)
                acc[i][j] = __builtin_amdgcn_wmma_f32_16x16x4_f32(
                    false, afrag[i], false, bfrag[j], (short)0, acc[i][j], false, false);
    }

    #pragma unroll
    for (int i = 0; i < 2; ++i) {
        #pragma unroll
        for (int j = 0; j < 2; ++j) {
            const int col = nBase + j * 16 + l16;
            const float bv = bias[col];
            #pragma unroll
            for (int v = 0; v < 8; ++v) {
                const int row = mBase + i * 16 + v + 8 * lhalf;
                const float val = acc[i][j][v] + bv;
                for (int rr = 0; rr < repeat; ++rr)
                    C[((size_t)row * repeat + rr) * ldc + col] = val;
            }
        }
    }
}

// ---------------------------------------------------------------------------
// Sequential scan with async-LDS double-buffered staging of x_proj.
// Block = 256 threads = 128 channels (contiguous d range within one b).
// Lane pair shares a channel (32 states each, registers only). Chunk of 16
// timesteps = 16 x 512B coalesced rows copied via GLOBAL_LOAD_ASYNC_TO_LDS_B32
// (ASYNCcnt), overlapped with compute on the previous chunk.
// y written in-place over x_proj; h_last -> d_out tail.
// ---------------------------------------------------------------------------
__global__ void scan_kernel(const float* __restrict__ Amat, const float* __restrict__ Bmat,
                            const float* __restrict__ csum, const float* __restrict__ Dv,
                            float* __restrict__ xz, float* __restrict__ hlast) {
    __shared__ float xs[2][SCH][128];

    const int tid = threadIdx.x;
    const int chbase = blockIdx.x * 128;     // first channel of this block
    const int dl  = tid >> 1;                // local channel 0..127
    const int sc  = tid & 1;                 // state-half select
    const int ch  = chbase + dl;             // b*1024 + d
    const int d   = ch & 1023;
    const int b   = ch >> 10;
    const int d0  = chbase & 1023;           // block's first d
    const int sb  = sc * 32;

    float h[32], a[32], w[32], cs[32];
    #pragma unroll
    for (int i = 0; i < 32; ++i) {
        h[i]  = 0.f;
        a[i]  = Amat[(size_t)d * DS + sb + i];
        w[i]  = Bmat[(size_t)d * DS + sb + i];
        cs[i] = csum[sb + i];
    }
    const float dv = Dv[d];

    // block's slab of x_proj: rows t (stride 2*DM floats), cols d0..d0+127
    float* gblk = xz + ((size_t)b * LS) * (2 * DM) + d0;
    float* xrow = xz + ((size_t)b * LS) * (2 * DM) + d; // per-channel column

    // async copy of one 16x128 chunk into xs[bufIdx]; 8 copies per thread
#define ISSUE_CHUNK(cidx, bufIdx)                                                  \
    {                                                                              \
        const float* gsrc = gblk + (size_t)(cidx) * SCH * (2 * DM);                \
        _Pragma("unroll")                                                          \
        for (int k_ = 0; k_ < 8; ++k_) {                                           \
            int idx_ = tid + k_ * 256;       /* 0..2047 */                         \
            int row_ = idx_ >> 7;                                                  \
            int dd_  = idx_ & 127;                                                 \
            unsigned long long ga_ =                                               \
                (unsigned long long)(gsrc + (size_t)row_ * (2 * DM) + dd_);        \
            unsigned la_ = (unsigned)(unsigned long long)&xs[bufIdx][row_][dd_];   \
            asm volatile("global_load_async_to_lds_b32 %0, %1, off"                \
                         :: "v"(la_), "v"(ga_) : "memory");                        \
        }                                                                          \
    }

    ISSUE_CHUNK(0, 0)

    for (int c = 0; c < NCHUNK; ++c) {
        if (c + 1 < NCHUNK) {
            ISSUE_CHUNK(c + 1, (c + 1) & 1)
            // 8 newest copies may be in flight; the previous 8 (chunk c) done
            asm volatile("s_wait_asynccnt 0x8" ::: "memory");
        } else {
            asm volatile("s_wait_asynccnt 0x0" ::: "memory");
        }
        __syncthreads();   // chunk c visible to all waves

        const float* xchunk = &xs[c & 1][0][0];
        #pragma unroll
        for (int tt = 0; tt < SCH; ++tt) {
            const int t = c * SCH + tt;
            const float xt = xchunk[tt * 128 + dl];
            float y = 0.f;
            #pragma unroll
            for (int i = 0; i < 32; ++i) {
                h[i] = tanhf(fmaf(h[i], a[i], xt * w[i]));
                y = fmaf(h[i], cs[i], y);
            }
            y += __shfl_xor(y, 1, 32);
            if (sc == 0) xrow[(size_t)t * (2 * DM)] = fmaf(dv, xt, y);
        }
        __syncthreads();   // buffer xs[c&1] free for reuse by copy of chunk c+2
    }
#undef ISSUE_CHUNK

    #pragma unroll
    for (int i = 0; i < 32; ++i) hlast[(size_t)ch * DS + sb + i] = h[i];
}

// ---------------------------------------------------------------------------
// u[r,d] = y[r,d] * silu(gate[r,d]);  y in xz[:, :1024], gate in xz[:, 1024:]
// ---------------------------------------------------------------------------
__global__ void gate_kernel(const float* __restrict__ xz, float* __restrict__ u) {
    const size_t idx = (size_t)blockIdx.x * blockDim.x + threadIdx.x; // < 4096*1024
    const size_t r = idx >> 10;
    const size_t d = idx & 1023;
    const float y = xz[r * (2 * DM) + d];
    const float g = xz[r * (2 * DM) + DM + d];
    const float sig = 1.0f / (1.0f + __expf(-g));
    u[idx] = y * g * sig;
}

// ---------------------------------------------------------------------------
extern "C" void kernel_launch(void* const* d_in, const int* in_sizes, int n_in,
                              void* d_out, int out_size, void* d_ws, size_t ws_size,
                              hipStream_t stream) {
    const float* x     = (const float*)d_in[0];
    const float* w_in  = (const float*)d_in[1];
    const float* b_in  = (const float*)d_in[2];
    const float* Amat  = (const float*)d_in[3];
    const float* Bm    = (const float*)d_in[4];
    const float* Cm    = (const float*)d_in[5];
    const float* Dv    = (const float*)d_in[6];
    const float* w_out = (const float*)d_in[7];
    const float* b_out = (const float*)d_in[8];
    const float* gnorm = (const float*)d_in[9];
    float* out = (float*)d_out;

    char* ws = (char*)d_ws;
    float* csum  = (float*)(ws);                                   // 256 B
    float* xnorm = (float*)(ws + 4096);                            // 16 MiB (reused as u)
    float* xz    = (float*)(ws + 4096 + (size_t)ROWS * DM * 4);    // 32 MiB

    float* hlast = out + (size_t)BATCH * SEQ * DM; // tail of d_out

    csum_kernel<<<1, 64, 0, stream>>>(Cm, csum);
    rmsnorm_kernel<<<ROWS, 256, 0, stream>>>(x, gnorm, xnorm);
    // GEMM1: [4096 x 1024] @ w_in^T -> xz [4096 x 2048]
    wmma_gemm_kernel<<<dim3(2 * DM / 128, ROWS / 64), 256, 0, stream>>>(
        xnorm, w_in, b_in, xz, DM, 2 * DM, 1);
    scan_kernel<<<(2 * ROWS) / 256, 256, 0, stream>>>(Amat, Bm, csum, Dv, xz, hlast);
    gate_kernel<<<((size_t)ROWS * DM) / 256, 256, 0, stream>>>(xz, xnorm);
    // GEMM2: u @ w_out^T -> out, rows duplicated x2 (jnp.repeat fused)
    wmma_gemm_kernel<<<dim3(DM / 128, ROWS / 64), 256, 0, stream>>>(
        xnorm, w_out, b_out, out, DM, DM, 2);
}